// CDFL1HistogramLoss_81690277970500
// MI455X (gfx1250) — compile-verified
//
#include <hip/hip_runtime.h>
#include <hip/hip_bf16.h>

#define IN_W   512
#define OUT_W  256
#define BINS   256
#define NCH    96          // 48 pred + 48 target channel-images
#define NPAIR  48
#define WRAD   20          // sigmoid window radius in bins; tail < 1e-10

typedef __attribute__((ext_vector_type(2))) float v2f;
typedef __attribute__((ext_vector_type(8))) float v8f;

// ---------------------------------------------------------------------------
// Antialiased 2x bilinear downsample tap (JAX resize default antialias=True):
// separable [1,3,3,1]/8 filter centered at input coord 2y+0.5, with
// out-of-range taps dropped and weights renormalized (edge handling).
// ---------------------------------------------------------------------------
__device__ __forceinline__ float resize_px(const float* __restrict__ img, int y, int x) {
    const float w4[4] = {0.125f, 0.375f, 0.375f, 0.125f};
    int r0 = 2 * y - 1, c0 = 2 * x - 1;
    float wy[4], wx[4], sy = 0.f, sx = 0.f;
    int   ry[4], cx[4];
#pragma unroll
    for (int j = 0; j < 4; ++j) {
        int r = r0 + j;
        bool vr = (r >= 0) && (r < IN_W);
        ry[j] = vr ? r : 0;
        wy[j] = vr ? w4[j] : 0.f;
        sy += wy[j];
        int c = c0 + j;
        bool vc = (c >= 0) && (c < IN_W);
        cx[j] = vc ? c : 0;
        wx[j] = vc ? w4[j] : 0.f;
        sx += wx[j];
    }
    float acc = 0.f;
#pragma unroll
    for (int j = 0; j < 4; ++j) {
        const float* row = img + (size_t)ry[j] * IN_W;
        float rs = wx[0] * row[cx[0]] + wx[1] * row[cx[1]] +
                   wx[2] * row[cx[2]] + wx[3] * row[cx[3]];
        acc += wy[j] * rs;
    }
    return acc * __frcp_rn(sy * sx);
}

__device__ __forceinline__ float sigm(float u) {
    // sigmoid(u) = 1/(1+e^-u)
    return __fdividef(1.0f, 1.0f + __expf(-u));
}

// ---------------------------------------------------------------------------
// Fused resize + windowed soft-histogram.
// grid = 96 channels * 4 row-subblocks; block = 256 threads.
// Outputs (global, f32 atomics):
//   Mbuf[b*96+ch] : count of pixels whose "all-ones" region covers bin b
//                   (difference-array marker; suffix-summed later via WMMA)
//   Wbuf[b*96+ch] : windowed sigmoid partial sums
//   S0buf[ch]     : sum_p sigmoid(300*x_p)
// ---------------------------------------------------------------------------
__global__ __launch_bounds__(256) void hist_kernel(
        const float* __restrict__ pred, const float* __restrict__ target,
        float* __restrict__ Mbuf, float* __restrict__ Wbuf,
        float* __restrict__ S0buf) {
    __shared__ float Ml[BINS];
    __shared__ float Wl[BINS];
    __shared__ float s0sh;

    int bid = blockIdx.x;
    int ch  = bid >> 2;          // 0..95
    int sub = bid & 3;           // row quarter of the 256x256 image
    const float* img = (ch < NPAIR)
        ? (pred   + (size_t)ch          * IN_W * IN_W)
        : (target + (size_t)(ch - NPAIR) * IN_W * IN_W);

    int t = threadIdx.x;
    if (t < BINS) { Ml[t] = 0.f; Wl[t] = 0.f; }
    if (t == 0) s0sh = 0.f;
    __syncthreads();

    float s0 = 0.f;
    for (int idx = t; idx < 64 * OUT_W; idx += 256) {
        int y = (sub << 6) + (idx >> 8);
        int x = idx & (OUT_W - 1);
        float v = resize_px(img, y, x);         // in [0,1)
        s0 += sigm(300.f * v);

        float f = v * 256.f;
        // thresholds j (=b+1) with j <= f - WRAD: sigmoid ~= 1 exactly
        int jones = (int)floorf(f - (float)WRAD);
        if (jones > 0) atomicAdd(&Ml[jones - 1], 1.0f);   // covers bins 0..jones-1
        int jlo = (jones + 1 > 1) ? jones + 1 : 1;
        int jhi = (int)ceilf(f + (float)WRAD);
        if (jhi > 256) jhi = 256;
        for (int j = jlo; j <= jhi; ++j) {
            float u = v - (float)j * (1.0f / 256.0f);
            atomicAdd(&Wl[j - 1], sigm(300.f * u));
        }
    }
    atomicAdd(&s0sh, s0);
    __syncthreads();

    if (t < BINS) {
        atomicAdd(&Mbuf[t * NCH + ch], Ml[t]);
        atomicAdd(&Wbuf[t * NCH + ch], Wl[t]);
    }
    if (t == 0) atomicAdd(&S0buf[ch], s0sh);
}

// ---------------------------------------------------------------------------
// Suffix-sum over bins as an exact f32 triangular matmul via WMMA:
//   T = U * M + W,  U[b,i] = (i >= b),  M,W,T are [256 bins x 96 channels].
// One wave per 16x16 output tile; V_WMMA_F32_16X16X4_F32, 64 K-steps.
// A-layout (16x4 f32): lanes 0-15 hold K=0,1; lanes 16-31 hold K=2,3.
// B/C/D: N = lane&15; row split low/high lane half per ISA 7.12.2.
// ---------------------------------------------------------------------------
__global__ __launch_bounds__(32) void suffix_wmma_kernel(
        const float* __restrict__ Mbuf, const float* __restrict__ Wbuf,
        float* __restrict__ Tbuf) {
    int mt = blockIdx.x & 15;    // bin-row tile 0..15
    int nt = blockIdx.x >> 4;    // channel tile 0..5
    int lane = threadIdx.x;      // wave32
    int m  = lane & 15;
    int kh = lane >> 4;          // 0: K pair {0,1}; 1: K pair {2,3}
    int gm = mt * 16 + m;        // global output bin row (for A)
    int gn = nt * 16 + m;        // global channel col (for B)

    v8f acc = {0.f, 0.f, 0.f, 0.f, 0.f, 0.f, 0.f, 0.f};
    for (int k0 = 0; k0 < BINS; k0 += 4) {
        int ka = k0 + 2 * kh;
        v2f a, b;
        a.x = (ka     >= gm) ? 1.0f : 0.0f;   // U[gm, ka]
        a.y = (ka + 1 >= gm) ? 1.0f : 0.0f;   // U[gm, ka+1]
        b.x = Mbuf[(size_t)ka       * NCH + gn];
        b.y = Mbuf[(size_t)(ka + 1) * NCH + gn];
        acc = __builtin_amdgcn_wmma_f32_16x16x4_f32(
                  false, a, false, b, (short)0, acc, false, false);
    }
#pragma unroll
    for (int r = 0; r < 8; ++r) {
        int gb = mt * 16 + r + 8 * kh;        // D row = r (+8 for high lanes)
        Tbuf[(size_t)gb * NCH + gn] = acc[r] + Wbuf[(size_t)gb * NCH + gn];
    }
}

// ---------------------------------------------------------------------------
// Final loss: cdf[b] = (S0 - T[b]) / (S0 - T[255]); mean |cdf_p - cdf_t|
// over 256 bins, mean over 48 channel pairs. Single block.
// ---------------------------------------------------------------------------
__global__ __launch_bounds__(256) void loss_kernel(
        const float* __restrict__ Tbuf, const float* __restrict__ S0buf,
        float* __restrict__ out) {
    __shared__ float red[256];
    int t = threadIdx.x;
    float acc = 0.f;
    for (int it = t; it < NPAIR * BINS; it += 256) {
        int c = it >> 8;
        int b = it & (BINS - 1);
        float S0p = S0buf[c],         S0t = S0buf[c + NPAIR];
        float dp  = S0p - Tbuf[(size_t)(BINS - 1) * NCH + c];
        float dt  = S0t - Tbuf[(size_t)(BINS - 1) * NCH + c + NPAIR];
        float cp  = (S0p - Tbuf[(size_t)b * NCH + c])          / dp;
        float ct  = (S0t - Tbuf[(size_t)b * NCH + c + NPAIR])  / dt;
        acc += fabsf(cp - ct);
    }
    red[t] = acc;
    __syncthreads();
    for (int s = 128; s > 0; s >>= 1) {
        if (t < s) red[t] += red[t + s];
        __syncthreads();
    }
    if (t == 0) out[0] = red[0] * (1.0f / (float)(NPAIR * BINS));
}

extern "C" void kernel_launch(void* const* d_in, const int* in_sizes, int n_in,
                              void* d_out, int out_size, void* d_ws, size_t ws_size,
                              hipStream_t stream) {
    const float* pred   = (const float*)d_in[0];
    const float* target = (const float*)d_in[1];
    float* out = (float*)d_out;

    float* Mbuf  = (float*)d_ws;            // [256][96]
    float* Wbuf  = Mbuf + BINS * NCH;       // [256][96]
    float* S0buf = Wbuf + BINS * NCH;       // [96]
    float* Tbuf  = S0buf + NCH;             // [256][96]

    // zero the atomic accumulators every call (deterministic per launch)
    hipMemsetAsync(d_ws, 0, (size_t)(2 * BINS * NCH + NCH) * sizeof(float), stream);

    hist_kernel<<<NCH * 4, 256, 0, stream>>>(pred, target, Mbuf, Wbuf, S0buf);
    suffix_wmma_kernel<<<NCH, 32, 0, stream>>>(Mbuf, Wbuf, Tbuf);
    loss_kernel<<<1, 256, 0, stream>>>(Tbuf, S0buf, out);
}